// GATModel_25374666785260
// MI455X (gfx1250) — compile-verified
//
#include <hip/hip_runtime.h>
#include <hip/hip_bf16.h>
#include <math.h>

#define N_NODES 50000
#define HIDD 64
#define KDIM 256          // K is 256 for all three layer GEMMs
#define LDSB_STRIDE 260   // 256 + 4 pad -> lanes land on distinct LDS banks
#define NEG_SLOPE 0.2f

typedef float v2f __attribute__((ext_vector_type(2)));
typedef float v8f __attribute__((ext_vector_type(8)));

// ---------------------------------------------------------------------------
// Fill kernel (grid-stride)
// ---------------------------------------------------------------------------
__global__ void gat_fill(float* __restrict__ p, float v, long long n) {
  long long i = (long long)blockIdx.x * blockDim.x + threadIdx.x;
  long long stride = (long long)gridDim.x * blockDim.x;
  for (; i < n; i += stride) p[i] = v;
}

// ---------------------------------------------------------------------------
// WMMA FP32 GEMM:  C[M,Nout] = A[M,K=256] * W^T   (W row-major [Nout,256])
//
// Block = 256 threads (8 waves). Block tile = 128(M) x 64(N).
//  * B panel (64 W-rows x 256) is one contiguous 64KB chunk of W; staged into
//    LDS with coalesced float4 loads, rows padded to 260 floats (260%64==4)
//    so the 16 lanes of a B-fragment read 16 distinct banks.
//  * Each wave owns 16 rows and all 64 cols: 4 accumulators; one A fragment
//    (2 floats/lane from global) is reused across 4 WMMAs per K-step.
//
// Fragment layouts per CDNA5 ISA:
//  A (16x4):  lane L -> row (L&15); VGPR0/1 = K = (L>>4)*2 + {0,1}
//  B (4x16):  lane L -> col (L&15); VGPR0/1 = K = (L>>4)*2 + {0,1}
//  D (16x16): VGPR i: lanes 0-15 -> (M=i, N=lane), lanes 16-31 -> (M=i+8)
// ---------------------------------------------------------------------------
__global__ void __launch_bounds__(256)
gat_gemm_wmma_f32(const float* __restrict__ A, const float* __restrict__ W,
                  float* __restrict__ C, int Nout) {
  __shared__ float ldsB[64 * LDSB_STRIDE];

  const int tid   = threadIdx.x;
  const int wave  = tid >> 5;
  const int lane  = tid & 31;
  const int r     = lane & 15;
  const int kk    = (lane >> 4) << 1;       // 0 or 2
  const int tileN = blockIdx.y << 6;        // 64-col panel
  const int tileM = (blockIdx.x << 7) + (wave << 4);

  // ---- stage B panel: contiguous 64KB of W -> LDS (all 256 threads) ----
  const float* Wp = W + (size_t)tileN * KDIM;
  for (int idx = tid * 4; idx < 64 * KDIM; idx += 256 * 4) {
    int n = idx >> 8;        // idx / 256
    int c = idx & 255;       // idx % 256
    *(float4*)&ldsB[n * LDSB_STRIDE + c] = *(const float4*)(Wp + idx);
  }
  __syncthreads();

  if (tileM >= N_NODES) return;   // uniform per-wave exit after the barrier

  const float* arow = A + (size_t)(tileM + r) * KDIM + kk;
  const float* b0   = ldsB + (size_t)(0 * 16 + r) * LDSB_STRIDE + kk;
  const float* b1   = ldsB + (size_t)(1 * 16 + r) * LDSB_STRIDE + kk;
  const float* b2   = ldsB + (size_t)(2 * 16 + r) * LDSB_STRIDE + kk;
  const float* b3   = ldsB + (size_t)(3 * 16 + r) * LDSB_STRIDE + kk;

  v8f acc0 = {}, acc1 = {}, acc2 = {}, acc3 = {};
#pragma unroll 4
  for (int k = 0; k < KDIM; k += 4) {
    v2f a = *(const v2f*)(arow + k);        // one A fragment, reused 4x
    v2f f0 = *(const v2f*)(b0 + k);
    v2f f1 = *(const v2f*)(b1 + k);
    v2f f2 = *(const v2f*)(b2 + k);
    v2f f3 = *(const v2f*)(b3 + k);
    acc0 = __builtin_amdgcn_wmma_f32_16x16x4_f32(false, a, false, f0, (short)0, acc0, false, false);
    acc1 = __builtin_amdgcn_wmma_f32_16x16x4_f32(false, a, false, f1, (short)0, acc1, false, false);
    acc2 = __builtin_amdgcn_wmma_f32_16x16x4_f32(false, a, false, f2, (short)0, acc2, false, false);
    acc3 = __builtin_amdgcn_wmma_f32_16x16x4_f32(false, a, false, f3, (short)0, acc3, false, false);
  }

  const int rowoff = (lane >> 4) << 3;      // 0 or 8
#pragma unroll
  for (int i = 0; i < 8; ++i) {
    size_t rowbase = (size_t)(tileM + rowoff + i) * Nout + tileN + r;
    C[rowbase + 0]  = acc0[i];
    C[rowbase + 16] = acc1[i];
    C[rowbase + 32] = acc2[i];
    C[rowbase + 48] = acc3[i];
  }
}

// ---------------------------------------------------------------------------
// el[n,h] = <feat[n,h,:], al[h,:]> ; er likewise
// ---------------------------------------------------------------------------
__global__ void gat_attn_proj(const float* __restrict__ feat,
                              const float* __restrict__ al,
                              const float* __restrict__ ar,
                              float* __restrict__ el, float* __restrict__ er,
                              int H) {
  int t = blockIdx.x * blockDim.x + threadIdx.x;
  if (t >= N_NODES * H) return;
  int n = t / H, h = t - n * H;
  const float* f  = feat + ((size_t)n * H + h) * HIDD;
  const float* a1 = al + h * HIDD;
  const float* a2 = ar + h * HIDD;
  float s1 = 0.f, s2 = 0.f;
#pragma unroll 8
  for (int d = 0; d < HIDD; ++d) {
    float v = f[d];
    s1 += v * a1[d];
    s2 += v * a2[d];
  }
  el[t] = s1;
  er[t] = s2;
}

// float atomic max via the signed/unsigned int ordering trick (NaN-free data)
__device__ __forceinline__ void atomicMaxF32(float* addr, float val) {
  if (!(val < 0.f)) {
    atomicMax((int*)addr, __float_as_int(val));
  } else {
    atomicMin((unsigned int*)addr, __float_as_uint(val));
  }
}

__device__ __forceinline__ float leaky(float v) {
  return v > 0.f ? v : NEG_SLOPE * v;
}

// ---------------------------------------------------------------------------
// Per-dst running max of leaky_relu(el[src]+er[dst])
// ---------------------------------------------------------------------------
__global__ void gat_edge_max(const int* __restrict__ src, const int* __restrict__ dst,
                             const float* __restrict__ el, const float* __restrict__ er,
                             float* __restrict__ m, int E, int H) {
  int e = blockIdx.x * blockDim.x + threadIdx.x;
  if (e >= E) return;
  int a = src[e], b = dst[e];
  for (int h = 0; h < H; ++h) {
    float v = leaky(el[a * H + h] + er[b * H + h]);
    atomicMaxF32(&m[b * H + h], v);
  }
}

// ---------------------------------------------------------------------------
// ex = exp(e - m[dst]); s[dst] += ex
// ---------------------------------------------------------------------------
__global__ void gat_edge_expsum(const int* __restrict__ src, const int* __restrict__ dst,
                                const float* __restrict__ el, const float* __restrict__ er,
                                const float* __restrict__ m, float* __restrict__ s,
                                float* __restrict__ ex, int E, int H) {
  int e = blockIdx.x * blockDim.x + threadIdx.x;
  if (e >= E) return;
  int a = src[e], b = dst[e];
  for (int h = 0; h < H; ++h) {
    float v = leaky(el[a * H + h] + er[b * H + h]);
    float x = expf(v - m[b * H + h]);
    ex[e * H + h] = x;
    atomicAdd(&s[b * H + h], x);
  }
}

// ---------------------------------------------------------------------------
// rst[dst,h,:] += feat[src,h,:] * (ex/s[dst,h]); one thread per (e,h,4 floats)
// ---------------------------------------------------------------------------
__global__ void gat_edge_scatter(const int* __restrict__ src, const int* __restrict__ dst,
                                 const float* __restrict__ ex, const float* __restrict__ s,
                                 const float* __restrict__ feat, float* __restrict__ rst,
                                 int E, int H) {
  long long t = (long long)blockIdx.x * blockDim.x + threadIdx.x;
  long long total = (long long)E * H * (HIDD / 4);
  if (t >= total) return;
  int per_e = H * (HIDD / 4);
  int e   = (int)(t / per_e);
  int rem = (int)(t - (long long)e * per_e);
  int h   = rem / (HIDD / 4);
  int q   = (rem - h * (HIDD / 4)) * 4;
  int a = src[e], b = dst[e];
  float alpha = ex[e * H + h] / s[b * H + h];
  const float4 f = *(const float4*)(feat + ((size_t)a * H + h) * HIDD + q);
  float* out = rst + ((size_t)b * H + h) * HIDD + q;
  atomicAdd(out + 0, f.x * alpha);
  atomicAdd(out + 1, f.y * alpha);
  atomicAdd(out + 2, f.z * alpha);
  atomicAdd(out + 3, f.w * alpha);
}

// ---------------------------------------------------------------------------
// Layers 1/2 finalize: out = elu(rst + bias (+ identity residual))
// ---------------------------------------------------------------------------
__global__ void gat_finalize_elu(const float* __restrict__ rst,
                                 const float* __restrict__ resid,
                                 const float* __restrict__ bias,
                                 float* __restrict__ out, int HD, int has_res) {
  long long t = (long long)blockIdx.x * blockDim.x + threadIdx.x;
  if (t >= (long long)N_NODES * HD) return;
  int o = (int)(t % HD);
  float v = rst[t] + bias[o];
  if (has_res) v += resid[t];
  v = v > 0.f ? v : (expf(v) - 1.f);
  out[t] = v;
}

// ---------------------------------------------------------------------------
// Layer 3 finalize: out[n,d] = mean_h( rst[n,h,d] + res[n,h,d] + b[h,d] )
// ---------------------------------------------------------------------------
__global__ void gat_finalize_mean(const float* __restrict__ rst,
                                  const float* __restrict__ res,
                                  const float* __restrict__ bias,
                                  float* __restrict__ out, int H) {
  int t = blockIdx.x * blockDim.x + threadIdx.x;
  if (t >= N_NODES * HIDD) return;
  int n = t / HIDD, d = t - n * HIDD;
  float acc = 0.f;
  for (int h = 0; h < H; ++h) {
    int o = h * HIDD + d;
    size_t idx = (size_t)n * H * HIDD + o;
    acc += rst[idx] + res[idx] + bias[o];
  }
  out[t] = acc * (1.0f / H);
}

// ---------------------------------------------------------------------------
extern "C" void kernel_launch(void* const* d_in, const int* in_sizes, int n_in,
                              void* d_out, int out_size, void* d_ws, size_t ws_size,
                              hipStream_t stream) {
  const float* x    = (const float*)d_in[0];
  const int*   src  = (const int*)d_in[1];
  const int*   dst  = (const int*)d_in[2];
  const float* W1   = (const float*)d_in[3];
  const float* al1  = (const float*)d_in[4];
  const float* ar1  = (const float*)d_in[5];
  const float* b1   = (const float*)d_in[6];
  const float* W2   = (const float*)d_in[7];
  const float* al2  = (const float*)d_in[8];
  const float* ar2  = (const float*)d_in[9];
  const float* b2   = (const float*)d_in[10];
  const float* W3   = (const float*)d_in[11];
  const float* al3  = (const float*)d_in[12];
  const float* ar3  = (const float*)d_in[13];
  const float* b3   = (const float*)d_in[14];
  const float* rW3  = (const float*)d_in[15];
  float* out = (float*)d_out;
  const int E = in_sizes[1];

  // workspace layout (floats)
  float* ws = (float*)d_ws;
  size_t o = 0;
  float* feat = ws + o; o += (size_t)N_NODES * 384;   // per-layer W h
  float* rst  = ws + o; o += (size_t)N_NODES * 384;   // aggregation accumulator
  float* res  = ws + o; o += (size_t)N_NODES * 384;   // layer-3 linear residual
  float* hA   = ws + o; o += (size_t)N_NODES * 256;   // layer-1 output
  float* hB   = ws + o; o += (size_t)N_NODES * 256;   // layer-2 output
  float* el   = ws + o; o += (size_t)N_NODES * 6;
  float* er   = ws + o; o += (size_t)N_NODES * 6;
  float* mbuf = ws + o; o += (size_t)N_NODES * 6;
  float* sbuf = ws + o; o += (size_t)N_NODES * 6;
  float* exb  = ws + o; o += (size_t)E * 6;

  const int TB = 256;
  const float NEG_INF = -__builtin_huge_valf();

  auto run_layer = [&](const float* hin, const float* W,
                       const float* al_, const float* ar_, const float* bias,
                       int H, const float* resW, int res_mode /*0 none,1 id,2 lin*/,
                       int act, float* hout) {
    const int HD = H * HIDD;
    // 1) feat = hin @ W^T  (and linear residual for layer 3)
    dim3 gg((N_NODES + 127) / 128, HD / 64);
    gat_gemm_wmma_f32<<<gg, 256, 0, stream>>>(hin, W, feat, HD);
    if (res_mode == 2)
      gat_gemm_wmma_f32<<<gg, 256, 0, stream>>>(hin, resW, res, HD);
    // 2) attention coefficients
    int nh = N_NODES * H;
    gat_attn_proj<<<(nh + TB - 1) / TB, TB, 0, stream>>>(feat, al_, ar_, el, er, H);
    // 3) init reductions
    gat_fill<<<2048, TB, 0, stream>>>(mbuf, NEG_INF, (long long)nh);
    gat_fill<<<2048, TB, 0, stream>>>(sbuf, 0.f, (long long)nh);
    gat_fill<<<4096, TB, 0, stream>>>(rst, 0.f, (long long)N_NODES * HD);
    // 4) edge softmax (max, exp+sum)
    gat_edge_max<<<(E + TB - 1) / TB, TB, 0, stream>>>(src, dst, el, er, mbuf, E, H);
    gat_edge_expsum<<<(E + TB - 1) / TB, TB, 0, stream>>>(src, dst, el, er, mbuf, sbuf,
                                                          exb, E, H);
    // 5) weighted scatter-add of messages
    long long nscatter = (long long)E * H * (HIDD / 4);
    gat_edge_scatter<<<(unsigned)((nscatter + TB - 1) / TB), TB, 0, stream>>>(
        src, dst, exb, sbuf, feat, rst, E, H);
    // 6) finalize
    long long nelem = (long long)N_NODES * HD;
    if (act) {
      gat_finalize_elu<<<(unsigned)((nelem + TB - 1) / TB), TB, 0, stream>>>(
          rst, hin, bias, hout, HD, res_mode == 1 ? 1 : 0);
    } else {
      gat_finalize_mean<<<(N_NODES * HIDD + TB - 1) / TB, TB, 0, stream>>>(
          rst, res, bias, out, H);
    }
  };

  // Layer 1: 256 -> 4x64, no residual, ELU
  run_layer(x, W1, al1, ar1, b1, 4, nullptr, 0, 1, hA);
  // Layer 2: 256 -> 4x64, identity residual, ELU
  run_layer(hA, W2, al2, ar2, b2, 4, nullptr, 1, 1, hB);
  // Layer 3: 256 -> 6x64, linear residual, no act, head-mean into d_out
  run_layer(hB, W3, al3, ar3, b3, 6, rW3, 2, 0, nullptr);
}